// RelationalGraphConv_28630251995288
// MI455X (gfx1250) — compile-verified
//
#include <hip/hip_runtime.h>

// RGCN for MI455X (gfx1250).
// - counting-sort edges by relation (histogram -> prefix -> scatter)
// - basis-combine relation weights, writing them PRE-PACKED into the WMMA
//   B-operand register layout so the GEMM inner loop loads B with b128.
// - message GEMM: 32 edges/block, 8 waves, each wave two 16x16 f32 WMMA
//   accumulators sharing each B fragment; atomic f32 scatter-add to targets.
// - final kernel: fused self-loop WMMA GEMM + degree normalize + bias.

typedef float v2f_t __attribute__((ext_vector_type(2)));
typedef float v8f_t __attribute__((ext_vector_type(8)));

#define N_NODES  50000
#define N_EDGES  800000
#define IN_F     128
#define OUT_F    128
#define N_REL    40
#define N_BASES  30

__device__ __forceinline__ int imin_i(int a, int b) { return a < b ? a : b; }

// Packed B layout: for group g (8 K-values), khalf (lane>>4), column n, store
// the 4 floats a lane consumes across two K=4 WMMA steps:
//   { B[8g+2*khalf][n], B[8g+2*khalf+1][n], B[8g+4+2*khalf][n], B[8g+4+2*khalf+1][n] }
// flat index: (((g*2 + khalf) * 128) + n) * 4 + slot
__device__ __forceinline__ size_t pack_index(int k, int n) {
  const int g     = k >> 3;
  const int k7    = k & 7;
  const int khalf = (k7 >> 1) & 1;
  const int slot  = ((k7 >> 2) << 1) | (k7 & 1);
  return ((size_t)(((g << 1) + khalf) * OUT_F + n) << 2) + slot;
}

// ---------- 1) histogram of edge types ----------
__global__ void k_hist(const int* __restrict__ etype, int* __restrict__ counts) {
  int e = blockIdx.x * blockDim.x + threadIdx.x;
  if (e < N_EDGES) atomicAdd(&counts[etype[e]], 1);
}

// ---------- 2) prefix sums: edge offsets + 32-edge tile offsets ----------
__global__ void k_prefix(const int* __restrict__ counts, int* __restrict__ offsets,
                         int* __restrict__ tileOff) {
  if (threadIdx.x == 0) {
    int o = 0, t = 0;
    for (int r = 0; r < N_REL; ++r) {
      offsets[r] = o; tileOff[r] = t;
      o += counts[r];
      t += (counts[r] + 31) >> 5;
    }
    offsets[N_REL] = o; tileOff[N_REL] = t;
  }
}

// ---------- 3) counting-sort scatter of edge ids + in-degree ----------
__global__ void k_scatter(const int* __restrict__ etype, const int* __restrict__ eidx,
                          const int* __restrict__ offsets, int* __restrict__ cursors,
                          int* __restrict__ sorted, float* __restrict__ deg) {
  int e = blockIdx.x * blockDim.x + threadIdx.x;
  if (e < N_EDGES) {
    int r = etype[e];
    int p = offsets[r] + atomicAdd(&cursors[r], 1);
    sorted[p] = e;
    atomicAdd(&deg[eidx[N_EDGES + e]], 1.0f);
  }
}

// ---------- 4) basis combine, writing packed B layout ----------
__global__ void k_basis(const float* __restrict__ coeff, const float* __restrict__ weight,
                        float* __restrict__ relWp) {
  int r  = blockIdx.y;
  int io = blockIdx.x * blockDim.x + threadIdx.x;  // io = k*128 + n
  float a = 0.f;
#pragma unroll
  for (int b = 0; b < N_BASES; ++b)
    a = fmaf(coeff[r * N_BASES + b], weight[b * (IN_F * OUT_F) + io], a);
  const int k = io >> 7, n = io & 127;
  relWp[(size_t)r * (IN_F * OUT_F) + pack_index(k, n)] = a;
}

// ---------- 4b) pack self_weight into the same layout ----------
__global__ void k_pack_self(const float* __restrict__ selfW, float* __restrict__ selfWp) {
  int io = blockIdx.x * blockDim.x + threadIdx.x;
  const int k = io >> 7, n = io & 127;
  selfWp[pack_index(k, n)] = selfW[io];
}

// ---------- 5) message GEMM: 32 edges/block, dual 16x16 WMMA per wave ----------
__global__ __launch_bounds__(256) void k_msg(
    const float* __restrict__ x, const int* __restrict__ eidx,
    const int* __restrict__ sorted, const int* __restrict__ offsets,
    const int* __restrict__ tileOff, const float* __restrict__ relWp,
    float* __restrict__ acc) {
  __shared__ float sA[32 * IN_F];  // 16 KB
  __shared__ int   sTgt[32];

  const int tile = blockIdx.x;
  if (tile >= tileOff[N_REL]) return;  // uniform per block

  // binary search: relation r with tileOff[r] <= tile < tileOff[r+1]
  int lo = 0, hi = N_REL - 1;
  while (lo < hi) {
    int mid = (lo + hi + 1) >> 1;
    if (tileOff[mid] <= tile) lo = mid; else hi = mid - 1;
  }
  const int r     = lo;
  const int ebase = offsets[r] + ((tile - tileOff[r]) << 5);
  const int ecnt  = imin_i(32, offsets[r + 1] - ebase);

  // stage 32 gathered source rows (zeros for missing edges of a partial tile)
  const int tid = threadIdx.x;
  const int row = tid >> 3;         // 0..31
  const int cb  = (tid & 7) << 4;   // 16 floats per thread
  float4* d = (float4*)(&sA[row * IN_F + cb]);
  if (row < ecnt) {
    const int e   = sorted[ebase + row];
    const int src = eidx[e];
    const float4* xs = (const float4*)(x + (size_t)src * IN_F + cb);
    d[0] = xs[0]; d[1] = xs[1]; d[2] = xs[2]; d[3] = xs[3];
  } else {
    float4 z = make_float4(0.f, 0.f, 0.f, 0.f);
    d[0] = z; d[1] = z; d[2] = z; d[3] = z;
  }
  if (tid < 32) sTgt[tid] = (tid < ecnt) ? eidx[N_EDGES + sorted[ebase + tid]] : 0;
  __syncthreads();

  const int lane  = tid & 31;
  const int wave  = tid >> 5;
  const int m     = lane & 15;       // A row within sub-tile
  const int khalf = lane >> 4;       // K-pair select
  const int n     = (wave << 4) + (lane & 15);
  const float* Wp = relWp + (size_t)r * (IN_F * OUT_F);

  v8f_t c0 = {}, c1 = {};
#pragma unroll
  for (int g = 0; g < 16; ++g) {
    const float4 bq =
        *(const float4*)(Wp + ((size_t)(((g << 1) + khalf) * OUT_F + n) << 2));
    v2f_t b0; b0.x = bq.x; b0.y = bq.y;   // K = 8g+2khalf, +1
    v2f_t b1; b1.x = bq.z; b1.y = bq.w;   // K = 8g+4+2khalf, +1
    const int k0 = (g << 3) + (khalf << 1);
    v2f_t a00 = *(const v2f_t*)(&sA[m * IN_F + k0]);
    v2f_t a01 = *(const v2f_t*)(&sA[m * IN_F + k0 + 4]);
    v2f_t a10 = *(const v2f_t*)(&sA[(16 + m) * IN_F + k0]);
    v2f_t a11 = *(const v2f_t*)(&sA[(16 + m) * IN_F + k0 + 4]);
    c0 = __builtin_amdgcn_wmma_f32_16x16x4_f32(false, a00, false, b0, (short)0, c0, false, false);
    c1 = __builtin_amdgcn_wmma_f32_16x16x4_f32(false, a10, false, b0, (short)0, c1, false, false);
    c0 = __builtin_amdgcn_wmma_f32_16x16x4_f32(false, a01, false, b1, (short)0, c0, false, false);
    c1 = __builtin_amdgcn_wmma_f32_16x16x4_f32(false, a11, false, b1, (short)0, c1, false, false);
  }

  // scatter-add: C VGPR i holds (M = i + 8*khalf, N = lane&15)
#pragma unroll
  for (int i = 0; i < 8; ++i) {
    const int mm = i + (khalf << 3);
    if (mm      < ecnt) atomicAdd(&acc[(size_t)sTgt[mm]      * OUT_F + n], c0[i]);
    if (mm + 16 < ecnt) atomicAdd(&acc[(size_t)sTgt[mm + 16] * OUT_F + n], c1[i]);
  }
}

// ---------- 6) self-loop GEMM fused with normalize + bias ----------
__global__ __launch_bounds__(256) void k_final(
    const float* __restrict__ x, const float* __restrict__ selfWp,
    const float* __restrict__ bias, const float* __restrict__ acc,
    const float* __restrict__ deg, float* __restrict__ out) {
  __shared__ float sA[32 * IN_F];
  const int node0 = blockIdx.x << 5;
  const int tid   = threadIdx.x;
  const int row   = tid >> 3;
  const int cb    = (tid & 7) << 4;
  const int nd0   = node0 + row;

  float4* d = (float4*)(&sA[row * IN_F + cb]);
  if (nd0 < N_NODES) {
    const float4* xs = (const float4*)(x + (size_t)nd0 * IN_F + cb);
    d[0] = xs[0]; d[1] = xs[1]; d[2] = xs[2]; d[3] = xs[3];
  } else {
    float4 z = make_float4(0.f, 0.f, 0.f, 0.f);
    d[0] = z; d[1] = z; d[2] = z; d[3] = z;
  }
  __syncthreads();

  const int lane  = tid & 31;
  const int wave  = tid >> 5;
  const int m     = lane & 15;
  const int khalf = lane >> 4;
  const int n     = (wave << 4) + (lane & 15);

  v8f_t c0 = {}, c1 = {};
#pragma unroll
  for (int g = 0; g < 16; ++g) {
    const float4 bq =
        *(const float4*)(selfWp + ((size_t)(((g << 1) + khalf) * OUT_F + n) << 2));
    v2f_t b0; b0.x = bq.x; b0.y = bq.y;
    v2f_t b1; b1.x = bq.z; b1.y = bq.w;
    const int k0 = (g << 3) + (khalf << 1);
    v2f_t a00 = *(const v2f_t*)(&sA[m * IN_F + k0]);
    v2f_t a01 = *(const v2f_t*)(&sA[m * IN_F + k0 + 4]);
    v2f_t a10 = *(const v2f_t*)(&sA[(16 + m) * IN_F + k0]);
    v2f_t a11 = *(const v2f_t*)(&sA[(16 + m) * IN_F + k0 + 4]);
    c0 = __builtin_amdgcn_wmma_f32_16x16x4_f32(false, a00, false, b0, (short)0, c0, false, false);
    c1 = __builtin_amdgcn_wmma_f32_16x16x4_f32(false, a10, false, b0, (short)0, c1, false, false);
    c0 = __builtin_amdgcn_wmma_f32_16x16x4_f32(false, a01, false, b1, (short)0, c0, false, false);
    c1 = __builtin_amdgcn_wmma_f32_16x16x4_f32(false, a11, false, b1, (short)0, c1, false, false);
  }

  const float bn = bias[n];
#pragma unroll
  for (int i = 0; i < 8; ++i) {
    const int mm = i + (khalf << 3);
    const int ndA = node0 + mm;
    const int ndB = node0 + mm + 16;
    if (ndA < N_NODES) {
      float dg = deg[ndA]; dg = dg < 1.f ? 1.f : dg;
      out[(size_t)ndA * OUT_F + n] = acc[(size_t)ndA * OUT_F + n] / dg + c0[i] + bn;
    }
    if (ndB < N_NODES) {
      float dg = deg[ndB]; dg = dg < 1.f ? 1.f : dg;
      out[(size_t)ndB * OUT_F + n] = acc[(size_t)ndB * OUT_F + n] / dg + c1[i] + bn;
    }
  }
}

extern "C" void kernel_launch(void* const* d_in, const int* in_sizes, int n_in,
                              void* d_out, int out_size, void* d_ws, size_t ws_size,
                              hipStream_t stream) {
  (void)in_sizes; (void)n_in; (void)out_size; (void)ws_size;
  const float* x      = (const float*)d_in[0];
  const int*   eidx   = (const int*)d_in[1];   // [2, E]: row0 = src, row1 = tgt
  const int*   etype  = (const int*)d_in[2];   // [E]
  const float* weight = (const float*)d_in[4]; // [B, IN, OUT]
  const float* coeff  = (const float*)d_in[5]; // [R, B]
  const float* selfW  = (const float*)d_in[6]; // [IN, OUT]
  const float* bias   = (const float*)d_in[7]; // [OUT]
  float*       out    = (float*)d_out;

  // workspace layout (all sections 256B aligned)
  char* p = (char*)d_ws;
  constexpr size_t ACC_B  = (size_t)N_NODES * OUT_F * sizeof(float);           // 25,600,000
  constexpr size_t DEG_B  = (((size_t)N_NODES * sizeof(float)) + 255) & ~(size_t)255;
  constexpr size_t RELW_B = (size_t)N_REL * IN_F * OUT_F * sizeof(float);      // 2,621,440
  constexpr size_t SELF_B = (size_t)IN_F * OUT_F * sizeof(float);              // 65,536
  constexpr size_t SMALL  = 256;
  float* acc     = (float*)p; p += ACC_B;
  float* deg     = (float*)p; p += DEG_B;
  float* relWp   = (float*)p; p += RELW_B;
  float* selfWp  = (float*)p; p += SELF_B;
  int*   counts  = (int*)p;   p += SMALL;
  int*   cursors = (int*)p;   p += SMALL;
  int*   offsets = (int*)p;   p += SMALL;  // N_REL+1 ints
  int*   tileOff = (int*)p;   p += SMALL;  // N_REL+1 ints
  int*   sorted  = (int*)p;                // N_EDGES ints

  hipMemsetAsync(acc, 0, ACC_B + DEG_B, stream);   // acc + deg contiguous
  hipMemsetAsync(counts, 0, 2 * SMALL, stream);    // counts + cursors

  k_hist<<<(N_EDGES + 255) / 256, 256, 0, stream>>>(etype, counts);
  k_prefix<<<1, 32, 0, stream>>>(counts, offsets, tileOff);
  k_scatter<<<(N_EDGES + 255) / 256, 256, 0, stream>>>(etype, eidx, offsets,
                                                       cursors, sorted, deg);
  k_basis<<<dim3((IN_F * OUT_F) / 256, N_REL), 256, 0, stream>>>(coeff, weight, relWp);
  k_pack_self<<<(IN_F * OUT_F) / 256, 256, 0, stream>>>(selfW, selfWp);

  const int maxTiles = N_EDGES / 32 + N_REL;  // upper bound on total 32-edge tiles
  k_msg<<<maxTiles, 256, 0, stream>>>(x, eidx, sorted, offsets, tileOff, relWp, acc);
  k_final<<<(N_NODES + 31) / 32, 256, 0, stream>>>(x, selfWp, bias, acc, deg, out);
}